// VisionMamba_61744449847367
// MI455X (gfx1250) — compile-verified
//
#include <hip/hip_runtime.h>
#include <hip/hip_bf16.h>
#include <math.h>

#define EMBED 384
#define DEPTH 12
#define DSTATE 16
#define DCONV 4
#define DTRANK 24
#define L_TOK 1024
#define PATCH_DIM 4096
#define B_SZ 2
#define M_ROWS (B_SZ * L_TOK)         // 2048
#define XPROJ_N (DTRANK + 2 * DSTATE) // 56

typedef float v2f __attribute__((ext_vector_type(2)));
typedef float v8f __attribute__((ext_vector_type(8)));

__device__ __forceinline__ float silu_f(float x) { return x / (1.f + __expf(-x)); }
__device__ __forceinline__ float softplus_f(float x) {
  return x > 0.f ? x + log1pf(__expf(-x)) : log1pf(__expf(x));
}

// ---- CDNA5 f32 WMMA: D(16x16) = A(16x4) * B(4x16) + C --------------------
__device__ __forceinline__ v8f wmma4(v2f a, v2f b, v8f c) {
  return __builtin_amdgcn_wmma_f32_16x16x4_f32(false, a, false, b, (short)0, c,
                                               false, false);
}

// ===========================================================================
// Generic row-major WMMA GEMM.  One wave computes a 16(M) x 64(N) tile.
// mode 0: D = A@B   mode 1: D = softplus(A@B + bias[col])   mode 2: D += A@B
// - B loads use CLAMPED column indices (WMMA columns are independent, so
//   garbage in cols >= N cannot pollute valid ones); stores are guarded.
// - Software-pipelined: next K-step's A/B fragments are loaded while the
//   current step's 4 WMMAs run, so the per-wmma s_wait_loadcnt has a full
//   iteration of latency slack.  Prefetch index is clamped -> no branches,
//   EXEC stays all-ones for every v_wmma.
// ===========================================================================
__global__ void wmma_gemm_f32(const float* __restrict__ A, int lda,
                              const float* __restrict__ Bm, int ldb,
                              float* __restrict__ D, int ldd,
                              const float* __restrict__ bias,
                              int M, int N, int K, int mode) {
  const int lane = threadIdx.x & 31;
  int wave = blockIdx.x * (blockDim.x >> 5) + (threadIdx.x >> 5);
  const int tiles_n = (N + 63) >> 6;
  const int tiles_m = M >> 4;
  const bool active = wave < tiles_m * tiles_n;
  if (!active) wave = 0;
  const int m0 = (wave / tiles_n) << 4;
  const int n0 = (wave % tiles_n) << 6;
  const int arow = m0 + (lane & 15);
  const int klo = (lane >> 4) << 1; // 0 for lanes 0-15, 2 for lanes 16-31
  const int ncol = lane & 15;
  const int c0 = n0 + ncol;
  const int l0 = min(c0, N - 1),      l1 = min(c0 + 16, N - 1);
  const int l2 = min(c0 + 32, N - 1), l3 = min(c0 + 48, N - 1);

  const float* Ap = A + (size_t)arow * lda + klo;
  const float* Bp = Bm + (size_t)klo * ldb;

  v2f a, b0, b1, b2, b3;
  a = *(const v2f*)(Ap);
  { const float* r0 = Bp; const float* r1 = Bp + ldb;
    b0.x = r0[l0]; b0.y = r1[l0]; b1.x = r0[l1]; b1.y = r1[l1];
    b2.x = r0[l2]; b2.y = r1[l2]; b3.x = r0[l3]; b3.y = r1[l3]; }

  v8f acc0 = {}, acc1 = {}, acc2 = {}, acc3 = {};
#pragma unroll 2
  for (int k = 0; k < K; k += 4) {
    const int kn = (k + 4 < K) ? (k + 4) : k; // clamped prefetch, branchless
    v2f an = *(const v2f*)(Ap + kn);
    const float* r0 = Bp + (size_t)kn * ldb;
    const float* r1 = r0 + ldb;
    v2f n0v, n1v, n2v, n3v;
    n0v.x = r0[l0]; n0v.y = r1[l0]; n1v.x = r0[l1]; n1v.y = r1[l1];
    n2v.x = r0[l2]; n2v.y = r1[l2]; n3v.x = r0[l3]; n3v.y = r1[l3];
    acc0 = wmma4(a, b0, acc0);
    acc1 = wmma4(a, b1, acc1);
    acc2 = wmma4(a, b2, acc2);
    acc3 = wmma4(a, b3, acc3);
    a = an; b0 = n0v; b1 = n1v; b2 = n2v; b3 = n3v;
  }
  if (!active) return;
  const int rbase = m0 + ((lane >> 4) << 3); // C/D: lanes16-31 hold rows +8
#pragma unroll
  for (int v = 0; v < 8; ++v) {
    const int row = rbase + v;
    float vals[4] = {acc0[v], acc1[v], acc2[v], acc3[v]};
#pragma unroll
    for (int j = 0; j < 4; ++j) {
      const int col = n0 + j * 16 + ncol;
      if (col < N) {
        float r = vals[j];
        if (mode == 1)      r = softplus_f(r + bias[col]);
        else if (mode == 2) r += D[(size_t)row * ldd + col];
        D[(size_t)row * ldd + col] = r;
      }
    }
  }
}

// ===========================================================================
// Patch embedding GEMM: tok(2048x384) = gather(x)(2048x4096) @ patch_w.
// k -> (ph,pw,pd) = (k>>8, (k>>3)&31, k&7); l -> (gh,gw,gd).  Same software
// pipelining as the generic GEMM.
// ===========================================================================
__global__ void patch_gemm_kernel(const float* __restrict__ X,
                                  const float* __restrict__ Wp,
                                  float* __restrict__ tok) {
  const int lane = threadIdx.x & 31;
  const int wave = blockIdx.x * (blockDim.x >> 5) + (threadIdx.x >> 5);
  const int tiles_n = EMBED / 64; // 6
  const int m0 = (wave / tiles_n) << 4;
  const int n0 = (wave % tiles_n) << 6;
  const int m = m0 + (lane & 15);
  const int b = m >> 10, l = m & 1023;
  const int gh = l >> 6, gw = (l >> 3) & 7, gd = l & 7;
  const float* xb = X + ((((size_t)b * 256 + gh * 16) * 256 + gw * 32) * 64 + gd * 8);
  const int klo = (lane >> 4) << 1;
  const int ncol = lane & 15;
  const float* Wb = Wp + (size_t)klo * EMBED + n0 + ncol;

  auto gatherA = [&](int k) -> v2f {
    const int kk = k + klo;
    const int ph = kk >> 8, pw = (kk >> 3) & 31, pd = kk & 7;
    return *(const v2f*)(xb + (size_t)ph * 16384 + pw * 64 + pd);
  };

  v2f a = gatherA(0);
  v2f b0, b1, b2, b3;
  { const float* r0 = Wb; const float* r1 = Wb + EMBED;
    b0.x = r0[0];  b0.y = r1[0];  b1.x = r0[16]; b1.y = r1[16];
    b2.x = r0[32]; b2.y = r1[32]; b3.x = r0[48]; b3.y = r1[48]; }

  v8f acc0 = {}, acc1 = {}, acc2 = {}, acc3 = {};
#pragma unroll 2
  for (int k = 0; k < PATCH_DIM; k += 4) {
    const int kn = (k + 4 < PATCH_DIM) ? (k + 4) : k;
    v2f an = gatherA(kn);
    const float* r0 = Wb + (size_t)kn * EMBED;
    const float* r1 = r0 + EMBED;
    v2f n0v, n1v, n2v, n3v;
    n0v.x = r0[0];  n0v.y = r1[0];  n1v.x = r0[16]; n1v.y = r1[16];
    n2v.x = r0[32]; n2v.y = r1[32]; n3v.x = r0[48]; n3v.y = r1[48];
    acc0 = wmma4(a, b0, acc0);
    acc1 = wmma4(a, b1, acc1);
    acc2 = wmma4(a, b2, acc2);
    acc3 = wmma4(a, b3, acc3);
    a = an; b0 = n0v; b1 = n1v; b2 = n2v; b3 = n3v;
  }
  const int rbase = m0 + ((lane >> 4) << 3);
#pragma unroll
  for (int v = 0; v < 8; ++v) {
    const int row = rbase + v;
    tok[(size_t)row * EMBED + n0 + ncol]      = acc0[v];
    tok[(size_t)row * EMBED + n0 + 16 + ncol] = acc1[v];
    tok[(size_t)row * EMBED + n0 + 32 + ncol] = acc2[v];
    tok[(size_t)row * EMBED + n0 + 48 + ncol] = acc3[v];
  }
}

// ---- per-batch coordinate maxima (dmax, max|angle|, max|vel|, max vel) ----
__global__ void coord_max_kernel(const float* __restrict__ bp,
                                 const unsigned char* __restrict__ mask,
                                 float* __restrict__ mx) {
  const int b = blockIdx.x;
  const float rr = bp[b * 4 + 0], ar = bp[b * 4 + 1], vr = bp[b * 4 + 2];
  const bool mk = mask[b] != 0;
  float m0 = -1e30f, m1 = -1e30f, m2 = -1e30f, m3 = -1e30f;
  for (int l = threadIdx.x; l < L_TOK; l += 32) {
    const int gh = l >> 6, gw = (l >> 3) & 7, gd = l & 7;
    const float dist = gh * rr;
    const float ang = (gw - 4.f) * ar;
    const float vel = mk ? (gd - 4.f) * vr : gd * vr;
    m0 = fmaxf(m0, dist);
    m1 = fmaxf(m1, fabsf(ang));
    m2 = fmaxf(m2, fabsf(vel));
    m3 = fmaxf(m3, vel);
  }
  for (int off = 16; off; off >>= 1) {
    m0 = fmaxf(m0, __shfl_xor(m0, off, 32));
    m1 = fmaxf(m1, __shfl_xor(m1, off, 32));
    m2 = fmaxf(m2, __shfl_xor(m2, off, 32));
    m3 = fmaxf(m3, __shfl_xor(m3, off, 32));
  }
  if (threadIdx.x == 0) {
    mx[b * 4 + 0] = m0; mx[b * 4 + 1] = m1;
    mx[b * 4 + 2] = m2; mx[b * 4 + 3] = m3;
  }
}

// ---- tok += patch_b + sincos positional embedding -------------------------
__global__ void pos_embed_add_kernel(float* __restrict__ tok,
                                     const float* __restrict__ pb,
                                     const float* __restrict__ bp,
                                     const unsigned char* __restrict__ mask,
                                     const float* __restrict__ mx) {
  const int idx = blockIdx.x * blockDim.x + threadIdx.x;
  if (idx >= M_ROWS * EMBED) return;
  const int d = idx % EMBED;
  const int m = idx / EMBED;
  const int b = m >> 10, l = m & 1023;
  const int gh = l >> 6, gw = (l >> 3) & 7, gd = l & 7;
  const float rr = bp[b * 4 + 0], ar = bp[b * 4 + 1], vr = bp[b * 4 + 2];
  const bool mk = mask[b] != 0;
  const int j = d >> 7, i = d & 127;
  float c;
  if (j == 0) {
    const float dmax = mx[b * 4 + 0];
    c = gh * rr / ((dmax == 0.f) ? 1.f : dmax);
  } else if (j == 1) {
    const float amax = mx[b * 4 + 1];
    c = (gw - 4.f) * ar / ((amax == 0.f) ? 1.f : amax);
  } else {
    const float v = mk ? (gd - 4.f) * vr : gd * vr;
    const float vam = mx[b * 4 + 2], vm = mx[b * 4 + 3];
    const float den = mk ? ((vam == 0.f) ? 1.f : vam) : ((vm == 0.f) ? 1.f : vm);
    c = v / den;
  }
  const int i2 = i & 63;
  const float om = __expf(-(float)i2 * (9.21034037197618f / 64.f)); // 10000^(-i/64)
  const float ang = c * om;
  tok[idx] += pb[d] + ((i < 64) ? __sinf(ang) : __cosf(ang));
}

// ---- layernorm over EMBED=384; one 128-thread block per token -------------
__global__ void layernorm_kernel(const float* __restrict__ in,
                                 const float* __restrict__ w,
                                 const float* __restrict__ b,
                                 float* __restrict__ out) {
  const int row = blockIdx.x;
  const float* x = in + (size_t)row * EMBED;
  const int t = threadIdx.x;
  float v0 = x[t], v1 = x[t + 128], v2 = x[t + 256];
  __shared__ float red[4];
  float s = v0 + v1 + v2;
  for (int off = 16; off; off >>= 1) s += __shfl_xor(s, off, 32);
  if ((t & 31) == 0) red[t >> 5] = s;
  __syncthreads();
  const float mu = (red[0] + red[1] + red[2] + red[3]) * (1.f / EMBED);
  __syncthreads();
  const float d0 = v0 - mu, d1 = v1 - mu, d2 = v2 - mu;
  float q = d0 * d0 + d1 * d1 + d2 * d2;
  for (int off = 16; off; off >>= 1) q += __shfl_xor(q, off, 32);
  if ((t & 31) == 0) red[t >> 5] = q;
  __syncthreads();
  const float var = (red[0] + red[1] + red[2] + red[3]) * (1.f / EMBED);
  const float rstd = rsqrtf(var + 1e-5f);
  float* o = out + (size_t)row * EMBED;
  o[t]       = d0 * rstd * w[t]       + b[t];
  o[t + 128] = d1 * rstd * w[t + 128] + b[t + 128];
  o[t + 256] = d2 * rstd * w[t + 256] + b[t + 256];
}

// ---- depthwise causal conv (DCONV=4) + silu; xi = xz[:, :, :384] ----------
__global__ void conv_silu_kernel(const float* __restrict__ xz,
                                 const float* __restrict__ cw,
                                 const float* __restrict__ cb,
                                 float* __restrict__ xc) {
  const int idx = blockIdx.x * blockDim.x + threadIdx.x;
  if (idx >= M_ROWS * EMBED) return;
  const int d = idx % EMBED;
  const int m = idx / EMBED;
  const int t = m & 1023;
  const float* base = xz + (size_t)m * (2 * EMBED) + d;
  float acc = cb[d];
#pragma unroll
  for (int j = 0; j < DCONV; ++j) {
    const int tt = t - (DCONV - 1) + j;
    if (tt >= 0) acc += base[(long)(tt - t) * (2 * EMBED)] * cw[d * DCONV + j];
  }
  xc[idx] = silu_f(acc);
}

// ---- selective scan: thread = (b, d, n); 16 lanes reduce the DSTATE dim ---
__global__ void scan_kernel(const float* __restrict__ dt,
                            const float* __restrict__ dbl,
                            const float* __restrict__ xc,
                            const float* __restrict__ Alog,
                            float* __restrict__ ys) {
  const int tid = blockIdx.x * blockDim.x + threadIdx.x; // 0..12287
  const int n = tid & 15;
  const int bd = tid >> 4; // 0..767
  const int b = bd / EMBED;
  const int d = bd % EMBED;
  const float A = -__expf(Alog[d * DSTATE + n]);
  const float* dtp = dt + (size_t)b * L_TOK * EMBED + d;
  const float* up  = xc + (size_t)b * L_TOK * EMBED + d;
  const float* Bp  = dbl + (size_t)b * L_TOK * XPROJ_N + DTRANK + n;
  const float* Cp  = dbl + (size_t)b * L_TOK * XPROJ_N + DTRANK + DSTATE + n;
  float* yp = ys + (size_t)b * L_TOK * EMBED + d;
  float h = 0.f;
  for (int t = 0; t < L_TOK; ++t) {
    const float dtt = dtp[(size_t)t * EMBED];
    const float ut  = up[(size_t)t * EMBED];
    const float Bt  = Bp[(size_t)t * XPROJ_N];
    const float Ct  = Cp[(size_t)t * XPROJ_N];
    h = h * __expf(dtt * A) + dtt * ut * Bt;
    float p = h * Ct;
    p += __shfl_xor(p, 1, 32);
    p += __shfl_xor(p, 2, 32);
    p += __shfl_xor(p, 4, 32);
    p += __shfl_xor(p, 8, 32);
    if (n == 0) yp[(size_t)t * EMBED] = p;
  }
}

// ---- y = (ys + xc * Dskip) * silu(z); z = xz[:, :, 384:] ------------------
__global__ void gate_kernel(const float* __restrict__ ys,
                            const float* __restrict__ xc,
                            const float* __restrict__ xz,
                            const float* __restrict__ Dp,
                            float* __restrict__ yg) {
  const int idx = blockIdx.x * blockDim.x + threadIdx.x;
  if (idx >= M_ROWS * EMBED) return;
  const int d = idx % EMBED;
  const int m = idx / EMBED;
  const float z = xz[(size_t)m * (2 * EMBED) + EMBED + d];
  yg[idx] = (ys[idx] + xc[idx] * Dp[d]) * silu_f(z);
}

// ===========================================================================
extern "C" void kernel_launch(void* const* d_in, const int* in_sizes, int n_in,
                              void* d_out, int out_size, void* d_ws, size_t ws_size,
                              hipStream_t stream) {
  const float* x       = (const float*)d_in[0];
  const float* bp      = (const float*)d_in[1];
  const float* patch_w = (const float*)d_in[2];
  const float* patch_b = (const float*)d_in[3];
  const float* ipw     = (const float*)d_in[4];
  const float* cw      = (const float*)d_in[5];
  const float* cb      = (const float*)d_in[6];
  const float* xpw     = (const float*)d_in[7];
  const float* dtw     = (const float*)d_in[8];
  const float* dtb     = (const float*)d_in[9];
  const float* Alog    = (const float*)d_in[10];
  const float* Dsk     = (const float*)d_in[11];
  const float* opw     = (const float*)d_in[12];
  const float* nw      = (const float*)d_in[13];
  const float* nb      = (const float*)d_in[14];
  const float* fnw     = (const float*)d_in[15];
  const float* fnb     = (const float*)d_in[16];
  const unsigned char* msk = (const unsigned char*)d_in[17];
  float* out = (float*)d_out;

  float* ws  = (float*)d_ws;
  float* tok = ws; ws += (size_t)M_ROWS * EMBED;
  float* hN  = ws; ws += (size_t)M_ROWS * EMBED;
  float* xz  = ws; ws += (size_t)M_ROWS * 2 * EMBED;
  float* xc  = ws; ws += (size_t)M_ROWS * EMBED;
  float* dbl = ws; ws += (size_t)M_ROWS * XPROJ_N;
  float* dt  = ws; ws += (size_t)M_ROWS * EMBED;
  float* ys  = ws; ws += (size_t)M_ROWS * EMBED;
  float* yg  = ws; ws += (size_t)M_ROWS * EMBED;
  float* mx  = ws; ws += 4 * B_SZ;

  const int ELEMS = M_ROWS * EMBED;
  const int EB = (ELEMS + 255) / 256;

  // tok = patch GEMM (2048x4096 @ 4096x384): 128*6 = 768 waves = 96 blocks
  patch_gemm_kernel<<<96, 256, 0, stream>>>(x, patch_w, tok);
  coord_max_kernel<<<B_SZ, 32, 0, stream>>>(bp, msk, mx);
  pos_embed_add_kernel<<<EB, 256, 0, stream>>>(tok, patch_b, bp, msk, mx);

  for (int lyr = 0; lyr < DEPTH; ++lyr) {
    layernorm_kernel<<<M_ROWS, 128, 0, stream>>>(
        tok, nw + (size_t)lyr * EMBED, nb + (size_t)lyr * EMBED, hN);
    // xz = hN @ in_proj (384x768): 128*12 = 1536 waves = 192 blocks
    wmma_gemm_f32<<<192, 256, 0, stream>>>(
        hN, EMBED, ipw + (size_t)lyr * EMBED * 2 * EMBED, 2 * EMBED,
        xz, 2 * EMBED, nullptr, M_ROWS, 2 * EMBED, EMBED, 0);
    conv_silu_kernel<<<EB, 256, 0, stream>>>(
        xz, cw + (size_t)lyr * EMBED * DCONV, cb + (size_t)lyr * EMBED, xc);
    // dbl = xc @ x_proj (384x56): 128 waves = 16 blocks
    wmma_gemm_f32<<<16, 256, 0, stream>>>(
        xc, EMBED, xpw + (size_t)lyr * EMBED * XPROJ_N, XPROJ_N,
        dbl, XPROJ_N, nullptr, M_ROWS, XPROJ_N, EMBED, 0);
    // dt = softplus(dbl[:, :24] @ dt_proj + dtb): K=24, 768 waves = 96 blocks
    wmma_gemm_f32<<<96, 256, 0, stream>>>(
        dbl, XPROJ_N, dtw + (size_t)lyr * DTRANK * EMBED, EMBED,
        dt, EMBED, dtb + (size_t)lyr * EMBED, M_ROWS, EMBED, DTRANK, 1);
    // selective scan: 12288 threads = 48 blocks
    scan_kernel<<<48, 256, 0, stream>>>(
        dt, dbl, xc, Alog + (size_t)lyr * EMBED * DSTATE, ys);
    gate_kernel<<<EB, 256, 0, stream>>>(ys, xc, xz, Dsk + (size_t)lyr * EMBED, yg);
    // tok += yg @ out_proj (384x384): 768 waves = 96 blocks, residual mode
    wmma_gemm_f32<<<96, 256, 0, stream>>>(
        yg, EMBED, opw + (size_t)lyr * EMBED * EMBED, EMBED,
        tok, EMBED, nullptr, M_ROWS, EMBED, EMBED, 2);
  }
  layernorm_kernel<<<M_ROWS, 128, 0, stream>>>(tok, fnw, fnb, out);
}